// NodeAutoencoder_40939628265722
// MI455X (gfx1250) — compile-verified
//
#include <hip/hip_runtime.h>
#include <math.h>

// ---------------- problem constants (match reference) ----------------
constexpr int BGR  = 8;
constexpr int NPER = 8192;
constexpr int DEG  = 16;
constexpr int NN   = BGR * NPER;     // 65536 nodes
constexpr int EPER = NPER * DEG;     // 131072
constexpr int EE   = BGR * EPER;     // 1,048,576 edges
constexpr int FD   = 64;             // IN == HID == OUT == 64
constexpr int KPER = NPER / 2;       // 4096 kept per graph
constexpr int KK   = BGR * KPER;     // 32768 pooled nodes

typedef float v2f __attribute__((ext_vector_type(2)));
typedef float v8f __attribute__((ext_vector_type(8)));

// ---------------- generic fills ----------------
__global__ void fill_f32(float* p, float v, int n) {
  int t = blockIdx.x * blockDim.x + threadIdx.x;
  if (t < n) p[t] = v;
}
__global__ void fill_i32(int* p, int v, int n) {
  int t = blockIdx.x * blockDim.x + threadIdx.x;
  if (t < n) p[t] = v;
}

// ---------------- f32 GEMM via V_WMMA_F32_16X16X4_F32 ----------------
// H[M x 64] = X[M x 64] @ W[64 x 64].
// One wave computes a full 16x64 output strip with 4 independent accumulators
// (one per 16-col tile) so the 4 WMMAs per k-step have no RAW dependence.
// A frag (32-bit A 16x4): lane = 16*khalf + m holds K = 2*khalf + {0,1}.
// C/D frag (32-bit 16x16): lane group selects M base (0 or 8); vgpr v -> M = base+v.
__global__ __launch_bounds__(32)
void gemm64_wmma(const float* __restrict__ X, const float* __restrict__ W,
                 float* __restrict__ H, int M) {
  int mt    = blockIdx.x;           // 16-row strip
  int lane  = threadIdx.x;
  int m     = lane & 15;
  int khalf = lane >> 4;            // 0 or 1
  int row0  = mt * 16;

  v8f c0 = {}, c1 = {}, c2 = {}, c3 = {};
#pragma unroll
  for (int kb = 0; kb < 16; ++kb) {
    int k0 = kb * 4;
    const float* ap = X + (size_t)(row0 + m) * FD + k0 + 2 * khalf;
    v2f a; a.x = ap[0]; a.y = ap[1];
    const float* bp = W + (size_t)(k0 + 2 * khalf) * FD + m;
    v2f b0; b0.x = bp[0];  b0.y = bp[FD];
    v2f b1; b1.x = bp[16]; b1.y = bp[FD + 16];
    v2f b2; b2.x = bp[32]; b2.y = bp[FD + 32];
    v2f b3; b3.x = bp[48]; b3.y = bp[FD + 48];
    c0 = __builtin_amdgcn_wmma_f32_16x16x4_f32(false, a, false, b0, (short)0, c0, false, false);
    c1 = __builtin_amdgcn_wmma_f32_16x16x4_f32(false, a, false, b1, (short)0, c1, false, false);
    c2 = __builtin_amdgcn_wmma_f32_16x16x4_f32(false, a, false, b2, (short)0, c2, false, false);
    c3 = __builtin_amdgcn_wmma_f32_16x16x4_f32(false, a, false, b3, (short)0, c3, false, false);
  }
  float* hp = H + (size_t)(row0 + 8 * khalf) * FD + m;
#pragma unroll
  for (int v = 0; v < 8; ++v) {
    hp[(size_t)v * FD +  0] = c0[v];
    hp[(size_t)v * FD + 16] = c1[v];
    hp[(size_t)v * FD + 32] = c2[v];
    hp[(size_t)v * FD + 48] = c3[v];
  }
}

// ---------------- degree / normalization ----------------
__global__ void deg_kernel(const int* __restrict__ dst, float* deg, int nE) {
  int e = blockIdx.x * blockDim.x + threadIdx.x;
  if (e < nE) unsafeAtomicAdd(&deg[dst[e]], 1.0f);
}
__global__ void dis_kernel(const float* __restrict__ deg, float* dis, int n) {
  int i = blockIdx.x * blockDim.x + threadIdx.x;
  if (i < n) dis[i] = rsqrtf(deg[i] + 1.0f);   // +1 self loop
}

// agg[dst] += h[src] * dis[src]*dis[dst], one thread per (edge, 4 features):
// one b128 load + 4 no-return f32 atomics.
__global__ void scatter_norm(const int* __restrict__ src, const int* __restrict__ dst,
                             const float* __restrict__ dis, const float* __restrict__ h,
                             float* agg) {
  size_t t = (size_t)blockIdx.x * blockDim.x + threadIdx.x;
  if (t >= (size_t)EE * 16) return;
  int e = (int)(t >> 4), f = (int)(t & 15) * 4;
  int s = src[e], d = dst[e];
  float norm = dis[s] * dis[d];
  float4 hv = *(const float4*)(h + (size_t)s * FD + f);
  float* ap = agg + (size_t)d * FD + f;
  unsafeAtomicAdd(ap + 0, hv.x * norm);
  unsafeAtomicAdd(ap + 1, hv.y * norm);
  unsafeAtomicAdd(ap + 2, hv.z * norm);
  unsafeAtomicAdd(ap + 3, hv.w * norm);
}

// aggr[dst] += h[src]  (pooling score aggregation, weight 1)
__global__ void scatter_plain(const int* __restrict__ src, const int* __restrict__ dst,
                              const float* __restrict__ h, float* agg) {
  size_t t = (size_t)blockIdx.x * blockDim.x + threadIdx.x;
  if (t >= (size_t)EE * 16) return;
  int e = (int)(t >> 4), f = (int)(t & 15) * 4;
  float4 hv = *(const float4*)(h + (size_t)src[e] * FD + f);
  float* ap = agg + (size_t)dst[e] * FD + f;
  unsafeAtomicAdd(ap + 0, hv.x);
  unsafeAtomicAdd(ap + 1, hv.y);
  unsafeAtomicAdd(ap + 2, hv.z);
  unsafeAtomicAdd(ap + 3, hv.w);
}

// out = relu(agg + H*dis^2 + bias), vectorized by 4
__global__ void slrelu_kernel(const float* __restrict__ agg, const float* __restrict__ H,
                              const float* __restrict__ dis, const float* __restrict__ bias,
                              float* out, int n) {
  size_t t = (size_t)blockIdx.x * blockDim.x + threadIdx.x;
  if (t >= (size_t)n * 16) return;
  int i = (int)(t >> 4), f = (int)(t & 15) * 4;
  float d = dis[i];
  float d2 = d * d;
  float4 av = *(const float4*)(agg + (size_t)i * FD + f);
  float4 hv = *(const float4*)(H + (size_t)i * FD + f);
  float4 bv = *(const float4*)(bias + f);
  float4 r;
  r.x = fmaxf(av.x + hv.x * d2 + bv.x, 0.f);
  r.y = fmaxf(av.y + hv.y * d2 + bv.y, 0.f);
  r.z = fmaxf(av.z + hv.z * d2 + bv.z, 0.f);
  r.w = fmaxf(av.w + hv.w * d2 + bv.w, 0.f);
  *(float4*)(out + (size_t)i * FD + f) = r;
}

// score[i] = tanh(dot(aggr2_i, Wl) + bl + dot(h1_i, Wr)), one wave per node
__global__ __launch_bounds__(256)
void score_kernel(const float* __restrict__ h1, const float* __restrict__ aggr2,
                  const float* __restrict__ Wl, const float* __restrict__ bl,
                  const float* __restrict__ Wr, float* score) {
  int wv   = (int)((blockIdx.x * (size_t)blockDim.x + threadIdx.x) >> 5);
  int lane = threadIdx.x & 31;
  if (wv >= NN) return;
  const float* hp = h1 + (size_t)wv * FD;
  const float* ap = aggr2 + (size_t)wv * FD;
  float p = ap[lane] * Wl[lane] + ap[lane + 32] * Wl[lane + 32]
          + hp[lane] * Wr[lane] + hp[lane + 32] * Wr[lane + 32];
#pragma unroll
  for (int off = 16; off > 0; off >>= 1) p += __shfl_xor(p, off, 32);
  if (lane == 0) score[wv] = tanhf(p + bl[0]);
}

// ---------------- per-graph exact top-k (bitonic sort in LDS) ----------------
// Order: value descending, index ascending (matches jax.lax.top_k).
__global__ __launch_bounds__(1024)
void topk_kernel(const float* __restrict__ score, float* sv, int* perm, int* new_idx) {
  __shared__ float sval[NPER];
  __shared__ int   sidx[NPER];
  int g = blockIdx.x, tid = threadIdx.x;
  for (int i = tid; i < NPER; i += 1024) { sval[i] = score[g * NPER + i]; sidx[i] = i; }
  __syncthreads();
  for (int k = 2; k <= NPER; k <<= 1) {
    for (int j = k >> 1; j > 0; j >>= 1) {
      for (int i = tid; i < NPER; i += 1024) {
        int x = i ^ j;
        if (x > i) {
          float v0 = sval[i], v1 = sval[x];
          int   i0 = sidx[i], i1 = sidx[x];
          bool before = (v0 > v1) || (v0 == v1 && i0 < i1); // i-elem precedes x-elem
          bool dir = ((i & k) == 0);
          if (before != dir) { sval[i] = v1; sval[x] = v0; sidx[i] = i1; sidx[x] = i0; }
        }
      }
      __syncthreads();
    }
  }
  for (int r = tid; r < KPER; r += 1024) {
    int local = sidx[r];
    int gn = g * NPER + local;      // original node id
    int p  = g * KPER + r;          // pooled node id (rank order)
    perm[p] = gn; sv[p] = sval[r]; new_idx[gn] = p;
  }
}

// xp[p] = h1[perm[p]] * sv[p], vectorized by 4
__global__ void gather_scale(const float* __restrict__ h1, const int* __restrict__ perm,
                             const float* __restrict__ sv, float* xp) {
  size_t t = (size_t)blockIdx.x * blockDim.x + threadIdx.x;
  if (t >= (size_t)KK * 16) return;
  int p = (int)(t >> 4), f = (int)(t & 15) * 4;
  float s = sv[p];
  float4 hv = *(const float4*)(h1 + (size_t)perm[p] * FD + f);
  float4 r; r.x = hv.x * s; r.y = hv.y * s; r.z = hv.z * s; r.w = hv.w * s;
  *(float4*)(xp + (size_t)p * FD + f) = r;
}

// remap edges; dropped edges -> node 0 with weight 0
__global__ void remap_kernel(const int* __restrict__ src, const int* __restrict__ dst,
                             const int* __restrict__ new_idx,
                             int* ns, int* nd, float* nw, float* deg2) {
  int e = blockIdx.x * blockDim.x + threadIdx.x;
  if (e >= EE) return;
  int s = new_idx[src[e]], d = new_idx[dst[e]];
  bool m = (s >= 0) && (d >= 0);
  int s2 = m ? s : 0, d2 = m ? d : 0;
  ns[e] = s2; nd[e] = d2; nw[e] = m ? 1.f : 0.f;
  if (m) unsafeAtomicAdd(&deg2[d2], 1.0f);
}

__global__ void scatter_norm_w(const int* __restrict__ ns, const int* __restrict__ nd,
                               const float* __restrict__ nw, const float* __restrict__ dis2,
                               const float* __restrict__ xp, float* agg2) {
  size_t t = (size_t)blockIdx.x * blockDim.x + threadIdx.x;
  if (t >= (size_t)EE * 16) return;
  int e = (int)(t >> 4), f = (int)(t & 15) * 4;
  if (nw[e] == 0.f) return;       // zero-weight edges contribute nothing
  int s = ns[e], d = nd[e];
  float norm = dis2[s] * dis2[d];
  float4 hv = *(const float4*)(xp + (size_t)s * FD + f);
  float* ap = agg2 + (size_t)d * FD + f;
  unsafeAtomicAdd(ap + 0, hv.x * norm);
  unsafeAtomicAdd(ap + 1, hv.y * norm);
  unsafeAtomicAdd(ap + 2, hv.z * norm);
  unsafeAtomicAdd(ap + 3, hv.w * norm);
}

// ---------------- host launcher ----------------
extern "C" void kernel_launch(void* const* d_in, const int* in_sizes, int n_in,
                              void* d_out, int out_size, void* d_ws, size_t ws_size,
                              hipStream_t stream) {
  const float* x   = (const float*)d_in[0];
  const int*   ei  = (const int*)  d_in[1];
  const float* W1  = (const float*)d_in[3];
  const float* b1  = (const float*)d_in[4];
  const float* Wl  = (const float*)d_in[5];
  const float* bl  = (const float*)d_in[6];
  const float* Wr  = (const float*)d_in[7];
  const float* W2  = (const float*)d_in[8];
  const float* b2  = (const float*)d_in[9];
  const int* src = ei;            // edge_index[0]
  const int* dst = ei + EE;       // edge_index[1]

  // workspace layout (floats first, then ints); buffers reused across phases
  float* fw = (float*)d_ws;
  size_t o = 0;
  float* bufA  = fw + o; o += (size_t)NN * FD;   // H0, later xp
  float* bufB  = fw + o; o += (size_t)NN * FD;   // agg -> aggr2 -> H2
  float* bufC  = fw + o; o += (size_t)NN * FD;   // h1 -> agg2
  float* deg   = fw + o; o += NN;
  float* dis   = fw + o; o += NN;
  float* score = fw + o; o += NN;
  float* sv    = fw + o; o += KK;
  float* deg2  = fw + o; o += KK;
  float* dis2  = fw + o; o += KK;
  float* nw    = fw + o; o += EE;
  int* iw = (int*)(fw + o);
  size_t oi = 0;
  int* new_idx = iw + oi; oi += NN;
  int* perm    = iw + oi; oi += KK;
  int* ns      = iw + oi; oi += EE;
  int* nd      = iw + oi; oi += EE;

  const int T = 256;
  auto blocks = [](size_t n, int t) { return (unsigned)((n + t - 1) / t); };
  const size_t NF = (size_t)NN * FD, KF = (size_t)KK * FD;
  const size_t E4 = (size_t)EE * 16, N4 = (size_t)NN * 16, K4 = (size_t)KK * 16;

  // ---- conv1 ----
  gemm64_wmma<<<NN / 16, 32, 0, stream>>>(x, W1, bufA, NN);               // H0
  fill_f32<<<blocks(NN, T), T, 0, stream>>>(deg, 0.f, NN);
  deg_kernel<<<blocks(EE, T), T, 0, stream>>>(dst, deg, EE);
  dis_kernel<<<blocks(NN, T), T, 0, stream>>>(deg, dis, NN);
  fill_f32<<<blocks(NF, T), T, 0, stream>>>(bufB, 0.f, (int)NF);          // agg
  scatter_norm<<<blocks(E4, T), T, 0, stream>>>(src, dst, dis, bufA, bufB);
  slrelu_kernel<<<blocks(N4, T), T, 0, stream>>>(bufB, bufA, dis, b1, bufC, NN); // h1

  // ---- pooling score ----
  fill_f32<<<blocks(NF, T), T, 0, stream>>>(bufB, 0.f, (int)NF);          // aggr2
  scatter_plain<<<blocks(E4, T), T, 0, stream>>>(src, dst, bufC, bufB);
  score_kernel<<<blocks((size_t)NN * 32, T), T, 0, stream>>>(bufC, bufB, Wl, bl, Wr, score);

  // ---- top-k + gather ----
  fill_i32<<<blocks(NN, T), T, 0, stream>>>(new_idx, -1, NN);
  topk_kernel<<<BGR, 1024, 0, stream>>>(score, sv, perm, new_idx);
  gather_scale<<<blocks(K4, T), T, 0, stream>>>(bufC, perm, sv, bufA);    // xp -> bufA

  // ---- edge remap + conv2 ----
  fill_f32<<<blocks(KK, T), T, 0, stream>>>(deg2, 0.f, KK);
  remap_kernel<<<blocks(EE, T), T, 0, stream>>>(src, dst, new_idx, ns, nd, nw, deg2);
  dis_kernel<<<blocks(KK, T), T, 0, stream>>>(deg2, dis2, KK);
  gemm64_wmma<<<KK / 16, 32, 0, stream>>>(bufA, W2, bufB, KK);            // H2 -> bufB
  fill_f32<<<blocks(KF, T), T, 0, stream>>>(bufC, 0.f, (int)KF);          // agg2
  scatter_norm_w<<<blocks(E4, T), T, 0, stream>>>(ns, nd, nw, dis2, bufA, bufC);
  slrelu_kernel<<<blocks(K4, T), T, 0, stream>>>(bufC, bufB, dis2, b2, (float*)d_out, KK);
}